// MedianBlur_1236950581827
// MI455X (gfx1250) — compile-verified
//
#include <hip/hip_runtime.h>
#include <hip/hip_bf16.h>
#include <stdint.h>

#ifndef __has_builtin
#define __has_builtin(x) 0
#endif

// ---------------- problem constants ----------------
#define IMG_H   1024
#define IMG_W   1024
#define NPLANES 24            // 8 * 3
#define TX      64            // output tile width per block
#define TY      32            // output tile height per block
#define LW      66            // staged tile width  (TX + 2 halo)
#define LH      34            // staged tile height (TY + 2 halo)
#define LS      68            // LDS row stride (padded, keeps 8B alignment)
#define NTHREADS 256

typedef __attribute__((address_space(1))) int gint;   // global int
typedef __attribute__((address_space(3))) int lint;   // LDS int

// ---- async global->LDS copy (gfx1250 ASYNCcnt path) with portable fallback ----
__device__ __forceinline__ void lds_fill_b32(const float* g, float* l) {
#if __has_builtin(__builtin_amdgcn_global_load_async_to_lds_b32)
    __builtin_amdgcn_global_load_async_to_lds_b32((gint*)g, (lint*)l,
                                                  /*offset=*/0, /*cpol=*/0);
#else
    *l = *g;   // global_load_b32 + ds_store_b32 fallback
#endif
}

__device__ __forceinline__ void wait_async_all() {
#if __has_builtin(__builtin_amdgcn_s_wait_asynccnt)
    __builtin_amdgcn_s_wait_asynccnt(0);
#else
    asm volatile("s_wait_asynccnt 0" ::: "memory");
#endif
}

// ---- min/max/median helpers: compiler maps to v_min3/v_max3/v_med3_f32 ----
__device__ __forceinline__ float med3f(float a, float b, float c) {
    return __builtin_amdgcn_fmed3f(a, b, c);
}
__device__ __forceinline__ float min3f(float a, float b, float c) {
    return fminf(fminf(a, b), c);
}
__device__ __forceinline__ float max3f(float a, float b, float c) {
    return fmaxf(fmaxf(a, b), c);
}
// sort a 3-element row triple into (lo, mid, hi): 3 instructions
__device__ __forceinline__ void sort3(float a, float b, float c,
                                      float& lo, float& mid, float& hi) {
    lo  = min3f(a, b, c);
    mid = med3f(a, b, c);
    hi  = max3f(a, b, c);
}

__global__ __launch_bounds__(NTHREADS)
void median3x3_kernel(const float* __restrict__ in, float* __restrict__ out) {
    __shared__ float tile[LH * LS];

    const int tid   = threadIdx.x;
    const int bx    = blockIdx.x;     // 0..15  (x tiles)
    const int by    = blockIdx.y;     // 0..31  (y tiles)
    const int plane = blockIdx.z;     // 0..23

    const float* src = in  + (size_t)plane * (IMG_H * IMG_W);
    float*       dst = out + (size_t)plane * (IMG_H * IMG_W);

    const int gx0 = bx * TX - 1;      // global col of tile col 0 (halo)
    const int gy0 = by * TY - 1;      // global row of tile row 0 (halo)

    // ---- stage haloed 66x34 tile into LDS; zero-fill out-of-bounds (padding) ----
    for (int i = tid; i < LH * LW; i += NTHREADS) {
        const int r  = i / LW;
        const int c  = i - r * LW;
        const int gy = gy0 + r;
        const int gx = gx0 + c;
        float* lptr  = &tile[r * LS + c];
        if ((unsigned)gy < (unsigned)IMG_H && (unsigned)gx < (unsigned)IMG_W) {
            lds_fill_b32(src + (size_t)gy * IMG_W + gx, lptr);
        } else {
            *lptr = 0.0f;             // ds_store_b32 (border zero padding)
        }
    }
    wait_async_all();
    __syncthreads();

    // ---- compute: each thread owns a 1-wide x 8-tall output strip ----
    const int lx = tid & (TX - 1);        // 0..63 : output col within tile
    const int ly = (tid >> 6) * 8;        // 0,8,16,24 : first output row
    const int ox = bx * TX + lx;
    const int oy = by * TY + ly;

    const float* t0 = &tile[ly * LS + lx];   // window top-left for first output

    float lo0, mid0, hi0, lo1, mid1, hi1;
    sort3(t0[0],      t0[1],      t0[2],      lo0, mid0, hi0);
    sort3(t0[LS + 0], t0[LS + 1], t0[LS + 2], lo1, mid1, hi1);

#pragma unroll
    for (int s = 0; s < 8; ++s) {
        const float* r2 = t0 + (s + 2) * LS;
        float lo2, mid2, hi2;
        sort3(r2[0], r2[1], r2[2], lo2, mid2, hi2);

        const float mn = max3f(lo0, lo1, lo2);    // v_max3_f32
        const float mx = min3f(hi0, hi1, hi2);    // v_min3_f32
        const float md = med3f(mid0, mid1, mid2); // v_med3_f32
        dst[(size_t)(oy + s) * IMG_W + ox] = med3f(mn, md, mx);

        lo0 = lo1; mid0 = mid1; hi0 = hi1;        // slide window down: reuse
        lo1 = lo2; mid1 = mid2; hi1 = hi2;        // two sorted rows of three
    }
}

extern "C" void kernel_launch(void* const* d_in, const int* in_sizes, int n_in,
                              void* d_out, int out_size, void* d_ws, size_t ws_size,
                              hipStream_t stream) {
    (void)in_sizes; (void)n_in; (void)d_ws; (void)ws_size; (void)out_size;
    const float* in  = (const float*)d_in[0];
    float*       out = (float*)d_out;

    dim3 grid(IMG_W / TX, IMG_H / TY, NPLANES);   // 16 x 32 x 24
    dim3 block(NTHREADS);
    median3x3_kernel<<<grid, block, 0, stream>>>(in, out);
}